// LaplacianSmoothLoss_37598143709401
// MI455X (gfx1250) — compile-verified
//
#include <hip/hip_runtime.h>

typedef __attribute__((ext_vector_type(2))) float v2f;
typedef __attribute__((ext_vector_type(8))) float v8f;

#define ROW     160
#define PLANE   (160*160)        // 25600
#define CHSTR   (160*160*160)    // 4,096,000 elements per (b,c) channel
#define NX4     40               // 160 / 4
#define TPB     256
#define NBLOCKS 3000
#define ITERS   8                // 3000*256*8 == 6*160*160*40 tasks exactly
#define NTOT    24576000.0f      // total elements; 3*NTOT = 73,728,000 (exact in f32)

struct __align__(16) F4 { float v[4]; };

__device__ __forceinline__ F4 ldf4(const float* p) { return *(const F4*)p; }
__device__ __forceinline__ F4 zf4() { F4 z; z.v[0]=0.f; z.v[1]=0.f; z.v[2]=0.f; z.v[3]=0.f; return z; }

// Wave-level sum of `acc` across all 32 lanes via V_WMMA_F32_16X16X4_F32.
// A[m][0]=acc[lane m], A[m][2]=acc[lane m+16] (K=1,3 zero); B = all-ones
// (layout independent). D[m][n] = acc[m]+acc[m+16] for every n. Sum of a
// lane's 8 D VGPRs gives the half-sum; xor-shuffle by 16 completes the wave
// total. EXEC must be all 1s at the call site (full blocks, no early exit).
__device__ __forceinline__ float wave_reduce_f32(float acc) {
  v2f a; a[0] = acc;  a[1] = 0.0f;
  v2f b; b[0] = 1.0f; b[1] = 1.0f;
  v8f c = {};
  c = __builtin_amdgcn_wmma_f32_16x16x4_f32(false, a, false, b, (short)0, c, false, false);
  float s = c[0] + c[1] + c[2] + c[3] + c[4] + c[5] + c[6] + c[7];
  s += __shfl_xor(s, 16, 32);
  return s;
}

// Full 256-thread block reduction; result valid on threadIdx.x == 0.
__device__ __forceinline__ float block_reduce_f32(float acc) {
  __shared__ float wpart[TPB / 32];
  float s = wave_reduce_f32(acc);
  const int lane = threadIdx.x & 31;
  const int wv   = threadIdx.x >> 5;
  if (lane == 0) wpart[wv] = s;
  __syncthreads();
  float t = 0.f;
  if (threadIdx.x == 0) {
    #pragma unroll
    for (int i = 0; i < TPB / 32; ++i) t += wpart[i];
  }
  return t;
}

// Stage 1: each thread handles ITERS float4-columns; accumulates Σ|lap_s| over
// s ∈ {1,2,4}; one f32 partial per block written to workspace.
__global__ void __launch_bounds__(TPB)
lap_partial_kernel(const float* __restrict__ flow, float* __restrict__ partial) {
  float acc = 0.f;
  const int sArr[3] = {1, 2, 4};

  // Do NOT unroll: one task's ~15 float4 loads in flight keeps VGPR pressure
  // low (no scratch spills); latency is hidden by the 8 waves per block.
  #pragma unroll 1
  for (int it = 0; it < ITERS; ++it) {
    const int task = (it * NBLOCKS + (int)blockIdx.x) * TPB + (int)threadIdx.x;
    const int x4 = task % NX4;
    int r        = task / NX4;
    const int y  = r % ROW;  r /= ROW;
    const int z  = r % ROW;
    const int bc = r / ROW;                       // 0..5 = (b*3 + c)

    const float* base = flow + (size_t)bc * CHSTR + (size_t)(z * ROW + y) * ROW;
    const int x0 = x4 * 4;

    // 12-float x-window covering x0-4 .. x0+7; edge float4s zeroed == zero pad
    // (strides <= 4, so OOB x only occurs inside the edge float4s). Every wave
    // contains an x-edge lane, so these stay as predicated selects.
    const F4 c4 = ldf4(base + x0);
    const F4 l4 = (x4 > 0)       ? ldf4(base + x0 - 4) : zf4();
    const F4 r4 = (x4 < NX4 - 1) ? ldf4(base + x0 + 4) : zf4();
    float wnd[12];
    #pragma unroll
    for (int i = 0; i < 4; ++i) { wnd[i] = l4.v[i]; wnd[4 + i] = c4.v[i]; wnd[8 + i] = r4.v[i]; }

    // y/z neighbors: interior (90.25% of volume, near-uniform per wave since a
    // wave spans <=2 y values) issues all 12 b128 loads unconditionally as one
    // clause; only boundary waves pay the per-load predication cost.
    F4 ym[3], yp[3], zm[3], zp[3];
    if (y >= 4 && y < ROW - 4 && z >= 4 && z < ROW - 4) {
      #pragma unroll
      for (int si = 0; si < 3; ++si) {
        const int s = sArr[si];
        ym[si] = ldf4(base + x0 - s * ROW);
        yp[si] = ldf4(base + x0 + s * ROW);
        zm[si] = ldf4(base + x0 - s * PLANE);
        zp[si] = ldf4(base + x0 + s * PLANE);
      }
    } else {
      #pragma unroll
      for (int si = 0; si < 3; ++si) {
        const int s = sArr[si];
        ym[si] = (y - s >= 0)  ? ldf4(base + x0 - s * ROW)   : zf4();
        yp[si] = (y + s < ROW) ? ldf4(base + x0 + s * ROW)   : zf4();
        zm[si] = (z - s >= 0)  ? ldf4(base + x0 - s * PLANE) : zf4();
        zp[si] = (z + s < ROW) ? ldf4(base + x0 + s * PLANE) : zf4();
      }
    }

    #pragma unroll
    for (int si = 0; si < 3; ++si) {
      const int s = sArr[si];
      #pragma unroll
      for (int i = 0; i < 4; ++i) {
        const float lap = wnd[4 + i - s] + wnd[4 + i + s]
                        + ym[si].v[i] + yp[si].v[i] + zm[si].v[i] + zp[si].v[i]
                        - 6.0f * wnd[4 + i];
        acc += fabsf(lap);
      }
    }
  }

  const float t = block_reduce_f32(acc);
  if (threadIdx.x == 0) partial[blockIdx.x] = t;
}

// Stage 2: single block reduces the per-block partials deterministically and
// applies the 1/(3N) normalization.
__global__ void __launch_bounds__(TPB)
lap_final_kernel(const float* __restrict__ partial, float* __restrict__ out) {
  float acc = 0.f;
  #pragma unroll 1
  for (int i = threadIdx.x; i < NBLOCKS; i += TPB) acc += partial[i];
  const float t = block_reduce_f32(acc);
  if (threadIdx.x == 0) out[0] = t * (1.0f / (3.0f * NTOT));
}

extern "C" void kernel_launch(void* const* d_in, const int* in_sizes, int n_in,
                              void* d_out, int out_size, void* d_ws, size_t ws_size,
                              hipStream_t stream) {
  const float* flow   = (const float*)d_in[0];
  float*       outp   = (float*)d_out;
  float*       partml = (float*)d_ws;   // needs NBLOCKS * 4 = 12 KB scratch

  lap_partial_kernel<<<dim3(NBLOCKS), dim3(TPB), 0, stream>>>(flow, partml);
  lap_final_kernel<<<dim3(1), dim3(TPB), 0, stream>>>(partml, outp);
}